// GNN_14139032338636
// MI455X (gfx1250) — compile-verified
//
#include <hip/hip_runtime.h>
#include <hip/hip_bf16.h>

typedef __attribute__((ext_vector_type(2))) float v2f;
typedef __attribute__((ext_vector_type(8))) float v8f;

// ---------------------------------------------------------------------------
// Degree / normalization: deg[i] = 1 (self loop) + #edges with dst==i,
// then dinv[i] = rsqrt(deg[i])  (deg >= 1 always, no zero check needed)
// ---------------------------------------------------------------------------
__global__ __launch_bounds__(256) void deg_init_kernel(float* __restrict__ deg, int n) {
    int i = blockIdx.x * 256 + threadIdx.x;
    if (i < n) deg[i] = 1.0f;
}

__global__ __launch_bounds__(256) void deg_accum_kernel(const int* __restrict__ dst,
                                                        float* __restrict__ deg, int nEdges) {
    int e = blockIdx.x * 256 + threadIdx.x;
    if (e < nEdges) atomicAdd(&deg[dst[e]], 1.0f);
}

__global__ __launch_bounds__(256) void dinv_kernel(float* __restrict__ deg, int n) {
    int i = blockIdx.x * 256 + threadIdx.x;
    if (i < n) deg[i] = rsqrtf(deg[i]);
}

// ---------------------------------------------------------------------------
// Aggregation agg = Â * fin   (feature width F, float4-vectorized)
// Pass 1: self-loop term initializes agg (doubles as the zero-fill):
//         agg[i] = dinv[i]^2 * fin[i]
// Pass 2: per-edge scatter: agg[d] += dinv[s]*dinv[d] * fin[s]
// Consecutive lanes cover consecutive float4s of one edge -> contiguous
// 32B/256B gathers and atomic bursts, all L2-resident (fin/agg <= 25.6MB).
// ---------------------------------------------------------------------------
template <int F>
__global__ __launch_bounds__(256) void selfloop_init_kernel(const float* __restrict__ fin,
                                                            const float* __restrict__ dinv,
                                                            float* __restrict__ agg, int n) {
    constexpr int FV = F / 4;
    int t = blockIdx.x * 256 + threadIdx.x;
    if (t >= n * FV) return;
    int i  = t / FV;
    int f4 = t % FV;
    float di = dinv[i];
    float w  = di * di;
    float4 v = ((const float4*)(fin + (size_t)i * F))[f4];
    float4 o;
    o.x = v.x * w; o.y = v.y * w; o.z = v.z * w; o.w = v.w * w;
    ((float4*)(agg + (size_t)i * F))[f4] = o;
}

template <int F>
__global__ __launch_bounds__(256) void edge_scatter_kernel(const int* __restrict__ srcIdx,
                                                           const int* __restrict__ dstIdx,
                                                           const float* __restrict__ dinv,
                                                           const float* __restrict__ fin,
                                                           float* __restrict__ agg, int nEdges) {
    constexpr int FV = F / 4;
    long long t = (long long)blockIdx.x * 256 + threadIdx.x;
    if (t >= (long long)nEdges * FV) return;
    int e  = (int)(t / FV);
    int f4 = (int)(t % FV);
    int s = srcIdx[e];
    int d = dstIdx[e];
    float norm = dinv[s] * dinv[d];
    float4 v = ((const float4*)(fin + (size_t)s * F))[f4];
    float* base = agg + (size_t)d * F + f4 * 4;
    atomicAdd(base + 0, norm * v.x);
    atomicAdd(base + 1, norm * v.y);
    atomicAdd(base + 2, norm * v.z);
    atomicAdd(base + 3, norm * v.w);
}

// ---------------------------------------------------------------------------
// Dense GEMM + bias (+ optional ReLU) via V_WMMA_F32_16X16X4_F32.
// out[nRows x F_OUT] = act(A[nRows x F_IN] * W[F_IN x F_OUT] + bias)
// One wave per 16-row tile; the wave keeps all F_OUT/16 accumulator tiles
// live (<= 7 tiles = 56 VGPRs) and sweeps K in steps of 4.
// W and bias are staged in LDS (max 28KB+448B << 320KB WGP LDS).
// ISA layouts (7.12.2):
//   A (16x4 f32):  lane L: M = L%16, VGPR j holds K = j + 2*(L/16)
//   B (4x16 f32):  lane L: N = L%16, VGPR j holds K = j + 2*(L/16)
//   C/D (16x16):   lane L: N = L%16, VGPR j holds M = j + 8*(L/16)
// nRows = 100000 is a multiple of 16 -> no partial tiles. Early exit is
// wave-uniform, so EXEC is all ones at every WMMA (ISA requirement).
// ---------------------------------------------------------------------------
template <int F_IN, int F_OUT, bool RELU>
__global__ __launch_bounds__(256) void gemm_bias_act_kernel(const float* __restrict__ A,
                                                            const float* __restrict__ W,
                                                            const float* __restrict__ bias,
                                                            float* __restrict__ out, int nRows) {
    constexpr int NT = F_OUT / 16;
    __shared__ float ldsW[F_IN * F_OUT];
    __shared__ float ldsB[F_OUT];
    for (int i = threadIdx.x; i < F_IN * F_OUT; i += 256) ldsW[i] = W[i];
    if (threadIdx.x < F_OUT) ldsB[threadIdx.x] = bias[threadIdx.x];
    __syncthreads();

    const int wave = threadIdx.x >> 5;
    const int lane = threadIdx.x & 31;
    const int tile = blockIdx.x * 8 + wave;
    const int row0 = tile * 16;
    if (row0 >= nRows) return;  // wave-uniform: EXEC stays all-ones below

    const int half = lane >> 4;   // 0 or 1
    const int m    = lane & 15;

    v8f acc[NT] = {};

    const float* arow = A + (size_t)(row0 + m) * F_IN;

    for (int kk = 0; kk < F_IN; kk += 4) {
        v2f a;
        a.x = arow[kk + 2 * half + 0];
        a.y = arow[kk + 2 * half + 1];
#pragma unroll
        for (int t = 0; t < NT; ++t) {
            v2f b;
            b.x = ldsW[(kk + 2 * half + 0) * F_OUT + t * 16 + m];
            b.y = ldsW[(kk + 2 * half + 1) * F_OUT + t * 16 + m];
            acc[t] = __builtin_amdgcn_wmma_f32_16x16x4_f32(
                /*neg_a=*/false, a, /*neg_b=*/false, b,
                /*c_mod=*/(short)0, acc[t], /*reuse_a=*/false, /*reuse_b=*/false);
        }
    }

#pragma unroll
    for (int t = 0; t < NT; ++t) {
        const int col = t * 16 + m;
        const float bc = ldsB[col];
#pragma unroll
        for (int j = 0; j < 8; ++j) {
            const int row = row0 + j + 8 * half;
            float v = acc[t][j] + bc;
            if (RELU) v = fmaxf(v, 0.0f);
            out[(size_t)row * F_OUT + col] = v;
        }
    }
}

// ---------------------------------------------------------------------------
// Orchestration.  Uses the identity  Â(XW) = (ÂX)W  to aggregate at the
// *narrower* width each layer:  L1 scatter at F=8, L2/L3 at F=64.
// ---------------------------------------------------------------------------
extern "C" void kernel_launch(void* const* d_in, const int* in_sizes, int n_in,
                              void* d_out, int out_size, void* d_ws, size_t ws_size,
                              hipStream_t stream) {
    const float* x  = (const float*)d_in[0];
    const int*   ei = (const int*)d_in[1];
    const float* W1 = (const float*)d_in[2];
    const float* b1 = (const float*)d_in[3];
    const float* W2 = (const float*)d_in[4];
    const float* b2 = (const float*)d_in[5];
    const float* W3 = (const float*)d_in[6];
    const float* b3 = (const float*)d_in[7];
    float* out = (float*)d_out;

    const int N = in_sizes[0] / 8;      // 100000
    const int E = in_sizes[1] / 2;      // 3200000
    const int* srcIdx = ei;             // edge_index[0]
    const int* dstIdx = ei + E;         // edge_index[1]

    char* ws = (char*)d_ws;
    float* dinv = (float*)ws;                                   // N floats (400KB)
    float* bufA = (float*)(ws + (size_t)(1u << 20));            // N*64 floats (25.6MB)
    float* bufB = (float*)(ws + (size_t)(1u << 20) + 26214400); // N*64 floats (25.6MB)

    const dim3 blk(256);
    auto cdiv = [](long long a, long long b) { return (unsigned)((a + b - 1) / b); };

    // degrees -> dinv (computed once, reused by all 3 layers)
    deg_init_kernel<<<cdiv(N, 256), blk, 0, stream>>>(dinv, N);
    deg_accum_kernel<<<cdiv(E, 256), blk, 0, stream>>>(dstIdx, dinv, E);
    dinv_kernel<<<cdiv(N, 256), blk, 0, stream>>>(dinv, N);

    const unsigned gemmBlocks = cdiv((long long)N / 16, 8);  // 782

    // ---- Layer 1: agg1 = Â x  (F=8)  ;  h1 = relu(agg1 @ W1 + b1) ----
    selfloop_init_kernel<8><<<cdiv((long long)N * 2, 256), blk, 0, stream>>>(x, dinv, bufB, N);
    edge_scatter_kernel<8><<<cdiv((long long)E * 2, 256), blk, 0, stream>>>(srcIdx, dstIdx, dinv, x, bufB, E);
    gemm_bias_act_kernel<8, 64, true><<<gemmBlocks, blk, 0, stream>>>(bufB, W1, b1, bufA, N);

    // ---- Layer 2: agg2 = Â h1 (F=64) ;  h2 = relu(agg2 @ W2 + b2) ----
    selfloop_init_kernel<64><<<cdiv((long long)N * 16, 256), blk, 0, stream>>>(bufA, dinv, bufB, N);
    edge_scatter_kernel<64><<<cdiv((long long)E * 16, 256), blk, 0, stream>>>(srcIdx, dstIdx, dinv, bufA, bufB, E);
    gemm_bias_act_kernel<64, 64, true><<<gemmBlocks, blk, 0, stream>>>(bufB, W2, b2, bufA, N);

    // ---- Layer 3: agg3 = Â h2 (F=64) ;  out = agg3 @ W3 + b3 ----
    selfloop_init_kernel<64><<<cdiv((long long)N * 16, 256), blk, 0, stream>>>(bufA, dinv, bufB, N);
    edge_scatter_kernel<64><<<cdiv((long long)E * 16, 256), blk, 0, stream>>>(srcIdx, dstIdx, dinv, bufA, bufB, E);
    gemm_bias_act_kernel<64, 112, false><<<gemmBlocks, blk, 0, stream>>>(bufB, W3, b3, out, N);
}